// SANGraphHead_64037962383829
// MI455X (gfx1250) — compile-verified
//
#include <hip/hip_runtime.h>
#include <math.h>
#include <stdint.h>

typedef float v2f __attribute__((ext_vector_type(2)));
typedef float v8f __attribute__((ext_vector_type(8)));

#define N_NODES 8192
#define DMODEL  128
#define NHEAD   4
#define DHEAD   32
#define FFDIM   256
#define TOPK    100
#define NGRAPH  64
#define LN_EPS  1e-5f
#define NSTRIPS (N_NODES / 128)

__device__ __forceinline__ v8f wmma4(v2f a, v2f b, v8f c) {
  // D = A(16x4 f32) * B(4x16 f32) + C(16x16 f32)
  return __builtin_amdgcn_wmma_f32_16x16x4_f32(false, a, false, b, (short)0, c,
                                               false, false);
}

// Async global->LDS copy of 16 bytes per lane (CDNA5, tracked by ASYNCcnt).
__device__ __forceinline__ void async_b128(unsigned lds_off, const float* g) {
  asm volatile("global_load_async_to_lds_b128 %0, %1, off"
               :: "v"(lds_off), "v"(g)
               : "memory");
}
__device__ __forceinline__ void wait_async_le16() {
  asm volatile("s_wait_asynccnt 0x10" ::: "memory");
}
__device__ __forceinline__ void wait_async_0() {
  asm volatile("s_wait_asynccnt 0x0" ::: "memory");
}

// ---------------------------------------------------------------------------
// Kernel 1: sim = x@x^T (streamed, WMMA f32) + exact per-row top-100 selection
// Block: 256 threads (8 waves) handles 16 rows. B strips (128 cols x 128 k)
// are double-buffered in LDS via async copies; inner WMMA loop is LDS-only.
// ---------------------------------------------------------------------------
__global__ __launch_bounds__(256) void sim_topk_kernel(
    const float* __restrict__ x, int* __restrict__ topk) {
  __shared__ float sA[16 * 132];        // row tile of x (k-major, padded)
  __shared__ float sB[2][128 * 132];    // double-buffered column strip
  __shared__ float strip[16 * 132];     // 16 rows x 128 cols of sim
  __shared__ float hval[16 * TOPK];     // per-row min-heaps (values)
  __shared__ int   hidx[16 * TOPK];     // per-row min-heaps (indices)
  __shared__ int   hcnt[16];

  const int tid  = threadIdx.x;
  const int lane = tid & 31;
  const int wave = tid >> 5;
  const int R0   = blockIdx.x * 16;
  const int mrow = lane & 15;
  const int hi   = lane >> 4;

  // Stage A rows into LDS.
  for (int i = tid; i < 16 * DMODEL; i += 256) {
    int r = i >> 7, c = i & 127;
    sA[r * 132 + c] = x[(size_t)(R0 + r) * DMODEL + c];
  }
  if (tid < 16) hcnt[tid] = 0;

  // Issue async stage of strip 0 into buffer 0 (16 float4 per thread,
  // fully coalesced: consecutive threads -> consecutive 16B of a column).
  {
    float* sbuf = &sB[0][0];
#pragma unroll
    for (int it = 0; it < 16; ++it) {
      int e   = it * 256 + tid;      // 0..4095 float4 elements of the strip
      int col = e >> 5;              // 0..127
      int kq  = e & 31;              // float4 index along k
      const float* g = x + (size_t)col * DMODEL + (size_t)kq * 4;
      unsigned lds = (unsigned)(uintptr_t)(sbuf + col * 132 + kq * 4);
      async_b128(lds, g);
    }
  }

  for (int s = 0; s < NSTRIPS; ++s) {
    // Prefetch next strip into the other buffer, then wait for current one.
    if (s + 1 < NSTRIPS) {
      float* sbuf = &sB[(s + 1) & 1][0];
      const float* xs = x + (size_t)(s + 1) * 128 * DMODEL;
#pragma unroll
      for (int it = 0; it < 16; ++it) {
        int e   = it * 256 + tid;
        int col = e >> 5;
        int kq  = e & 31;
        const float* g = xs + (size_t)col * DMODEL + (size_t)kq * 4;
        unsigned lds = (unsigned)(uintptr_t)(sbuf + col * 132 + kq * 4);
        async_b128(lds, g);
      }
      wait_async_le16();             // current strip's 16 copies complete
    } else {
      wait_async_0();
    }
    __syncthreads();                 // all waves' copies visible

    // 16x16 WMMA tile for this wave, LDS-only operands.
    v8f acc = {0.f, 0.f, 0.f, 0.f, 0.f, 0.f, 0.f, 0.f};
    const float* ap = &sA[mrow * 132];
    const float* bp = &sB[s & 1][(wave * 16 + mrow) * 132];
    for (int kk = 0; kk < DMODEL; kk += 4) {
      int ko = kk + 2 * hi;
      v2f a; a.x = ap[ko]; a.y = ap[ko + 1];
      v2f b; b.x = bp[ko]; b.y = bp[ko + 1];
      acc = wmma4(a, b, acc);
    }
    const int scol = wave * 16 + mrow;
#pragma unroll
    for (int i = 0; i < 8; ++i)
      strip[(i + 8 * hi) * 132 + scol] = acc[i];
    __syncthreads();

    // One serialized lane per row updates that row's top-100 min-heap.
    if ((tid & 15) == 0) {
      const int m = tid >> 4;
      const int rowg = R0 + m;
      float* hv = &hval[m * TOPK];
      int*   hx = &hidx[m * TOPK];
      int cnt = hcnt[m];
      const float* srow = &strip[m * 132];
      const int cg0 = s * 128;
      for (int c = 0; c < 128; ++c) {
        int cg = cg0 + c;
        if (cg == rowg) continue;             // diagonal = -1e10, never top-k
        float v = srow[c];
        if (cnt < TOPK) {
          int i = cnt++;
          hv[i] = v; hx[i] = cg;
          while (i > 0) {                     // sift-up
            int p = (i - 1) >> 1;
            if (hv[p] <= hv[i]) break;
            float tv = hv[p]; hv[p] = hv[i]; hv[i] = tv;
            int   ti = hx[p]; hx[p] = hx[i]; hx[i] = ti;
            i = p;
          }
        } else if (v > hv[0]) {
          hv[0] = v; hx[0] = cg;
          int i = 0;
          for (;;) {                          // sift-down
            int l = 2 * i + 1, r = l + 1, sm = i;
            if (l < TOPK && hv[l] < hv[sm]) sm = l;
            if (r < TOPK && hv[r] < hv[sm]) sm = r;
            if (sm == i) break;
            float tv = hv[sm]; hv[sm] = hv[i]; hv[i] = tv;
            int   ti = hx[sm]; hx[sm] = hx[i]; hx[i] = ti;
            i = sm;
          }
        }
      }
      hcnt[m] = cnt;
    }
    __syncthreads();
  }

  {
    const int m = tid >> 4;
    for (int j = tid & 15; j < TOPK; j += 16)
      topk[(size_t)(R0 + m) * TOPK + j] = hidx[m * TOPK + j];
  }
}

// ---------------------------------------------------------------------------
// Kernel 2: generic WMMA GEMM  Y = epilogue(X @ W^T + b)
// Epilogue options: +residual, ReLU, LayerNorm. Block = 16 rows, 8 waves.
// ---------------------------------------------------------------------------
__global__ __launch_bounds__(256) void gemm_ln_kernel(
    const float* __restrict__ X, const float* __restrict__ W,
    const float* __restrict__ bias, const float* __restrict__ res,
    const float* __restrict__ ln_g, const float* __restrict__ ln_b,
    float* __restrict__ Y, int Cin, int Cout, int doRelu) {
  extern __shared__ float smem[];
  float* sX = smem;                       // 16 * (Cin + 4)
  float* sY = smem + 16 * (Cin + 4);      // 16 * (Cout + 4)
  const int sxs = Cin + 4, sys = Cout + 4;
  const int tid  = threadIdx.x;
  const int lane = tid & 31;
  const int wave = tid >> 5;
  const int R0   = blockIdx.x * 16;
  const int mrow = lane & 15;
  const int hi   = lane >> 4;

  for (int i = tid; i < 16 * Cin; i += 256) {
    int r = i / Cin, c = i - r * Cin;
    sX[r * sxs + c] = X[(size_t)(R0 + r) * Cin + c];
  }
  __syncthreads();

  const int nct = Cout >> 4;
  for (int ct = wave; ct < nct; ct += 8) {
    const int C0 = ct * 16;
    v8f acc = {0.f, 0.f, 0.f, 0.f, 0.f, 0.f, 0.f, 0.f};
    const float*  ap = &sX[mrow * sxs];
    const float2* bp =
        reinterpret_cast<const float2*>(W + (size_t)(C0 + mrow) * Cin);
    for (int kk = 0; kk < Cin; kk += 4) {
      int ko = kk + 2 * hi;
      v2f a; a.x = ap[ko]; a.y = ap[ko + 1];
      float2 bf = bp[ko >> 1];
      v2f b; b.x = bf.x; b.y = bf.y;
      acc = wmma4(a, b, acc);
    }
    float bc = bias ? bias[C0 + mrow] : 0.f;
#pragma unroll
    for (int i = 0; i < 8; ++i)
      sY[(i + 8 * hi) * sys + C0 + mrow] = acc[i] + bc;
  }
  __syncthreads();

  if (res) {
    for (int i = tid; i < 16 * Cout; i += 256) {
      int r = i / Cout, c = i - r * Cout;
      sY[r * sys + c] += res[(size_t)(R0 + r) * Cout + c];
    }
    __syncthreads();
  }

  if (ln_g) {                       // LayerNorm: 16 lanes per row
    const int m = tid >> 4, j = tid & 15;
    float ssum = 0.f;
    for (int c = j; c < Cout; c += 16) ssum += sY[m * sys + c];
    ssum += __shfl_xor(ssum, 1); ssum += __shfl_xor(ssum, 2);
    ssum += __shfl_xor(ssum, 4); ssum += __shfl_xor(ssum, 8);
    float mean = ssum / (float)Cout;
    float vsum = 0.f;
    for (int c = j; c < Cout; c += 16) {
      float d = sY[m * sys + c] - mean; vsum += d * d;
    }
    vsum += __shfl_xor(vsum, 1); vsum += __shfl_xor(vsum, 2);
    vsum += __shfl_xor(vsum, 4); vsum += __shfl_xor(vsum, 8);
    float inv = rsqrtf(vsum / (float)Cout + LN_EPS);
    for (int c = j; c < Cout; c += 16) {
      float vv = (sY[m * sys + c] - mean) * inv * ln_g[c] + ln_b[c];
      Y[(size_t)(R0 + m) * Cout + c] = vv;
    }
  } else {
    for (int i = tid; i < 16 * Cout; i += 256) {
      int r = i / Cout, c = i - r * Cout;
      float vv = sY[r * sys + c];
      if (doRelu) vv = fmaxf(vv, 0.f);
      Y[(size_t)(R0 + r) * Cout + c] = vv;
    }
  }
}

// ---------------------------------------------------------------------------
// Kernel 3: sparse masked attention over {self} U top-100 neighbors.
// Block = 128 threads = 4 waves (one head per wave), one node per block.
// ---------------------------------------------------------------------------
__global__ __launch_bounds__(128) void attn_kernel(
    const float* __restrict__ qkv, const int* __restrict__ topk,
    float* __restrict__ ctx) {
  __shared__ int nidx[TOPK + 1];
  const int row  = blockIdx.x;
  const int tid  = threadIdx.x;
  const int lane = tid & 31;
  const int h    = tid >> 5;

  if (tid == 0) nidx[0] = row;
  if (tid >= 1 && tid <= TOPK) nidx[tid] = topk[(size_t)row * TOPK + tid - 1];
  __syncthreads();

  const float scale = 0.17677669529663687f;   // 1/sqrt(32)
  const float qd = qkv[(size_t)row * 384 + h * DHEAD + lane];
  float s0 = -INFINITY, s1 = -INFINITY, s2 = -INFINITY, s3 = -INFINITY;

  for (int j = 0; j <= TOPK; ++j) {
    int cj = nidx[j];
    float kd = qkv[(size_t)cj * 384 + DMODEL + h * DHEAD + lane];
    float p = qd * kd;
    p += __shfl_xor(p, 16); p += __shfl_xor(p, 8);
    p += __shfl_xor(p, 4);  p += __shfl_xor(p, 2); p += __shfl_xor(p, 1);
    p *= scale;
    if (lane == (j & 31)) {
      int slot = j >> 5;
      if (slot == 0) s0 = p; else if (slot == 1) s1 = p;
      else if (slot == 2) s2 = p; else s3 = p;
    }
  }
  float mx = fmaxf(fmaxf(s0, s1), fmaxf(s2, s3));
  mx = fmaxf(mx, __shfl_xor(mx, 16)); mx = fmaxf(mx, __shfl_xor(mx, 8));
  mx = fmaxf(mx, __shfl_xor(mx, 4));  mx = fmaxf(mx, __shfl_xor(mx, 2));
  mx = fmaxf(mx, __shfl_xor(mx, 1));
  float e0 = expf(s0 - mx), e1 = expf(s1 - mx);
  float e2 = expf(s2 - mx), e3 = expf(s3 - mx);
  float sum = e0 + e1 + e2 + e3;
  sum += __shfl_xor(sum, 16); sum += __shfl_xor(sum, 8);
  sum += __shfl_xor(sum, 4);  sum += __shfl_xor(sum, 2); sum += __shfl_xor(sum, 1);
  float rs = 1.f / sum;
  e0 *= rs; e1 *= rs; e2 *= rs; e3 *= rs;

  float cacc = 0.f;
  for (int j = 0; j <= TOPK; ++j) {
    int slot = j >> 5;
    float av = (slot == 0) ? e0 : (slot == 1) ? e1 : (slot == 2) ? e2 : e3;
    float aj = __shfl(av, j & 31);
    int cj = nidx[j];
    float vd = qkv[(size_t)cj * 384 + 2 * DMODEL + h * DHEAD + lane];
    cacc += aj * vd;
  }
  ctx[(size_t)row * DMODEL + h * DHEAD + lane] = cacc;
}

// ---------------------------------------------------------------------------
// Pooling + head MLP
// ---------------------------------------------------------------------------
__global__ void pool_init_kernel(float* sums, float* cnt) {
  int i = blockIdx.x * blockDim.x + threadIdx.x;
  if (i < NGRAPH * DMODEL) sums[i] = 0.f;
  if (i < NGRAPH) cnt[i] = 0.f;
}

__global__ __launch_bounds__(128) void pool_kernel(
    const float* __restrict__ h2, const int* __restrict__ bidx,
    float* __restrict__ sums, float* __restrict__ cnt) {
  int n = blockIdx.x;
  int g = bidx[n];
  atomicAdd(&sums[g * DMODEL + threadIdx.x],
            h2[(size_t)n * DMODEL + threadIdx.x]);
  if (threadIdx.x == 0) atomicAdd(&cnt[g], 1.f);
}

__global__ __launch_bounds__(128) void head_kernel(
    const float* __restrict__ sums, const float* __restrict__ cnt,
    const float* __restrict__ Wm1, const float* __restrict__ bm1,
    const float* __restrict__ Wm2, const float* __restrict__ bm2,
    float* __restrict__ out) {
  __shared__ float sp[DMODEL];
  __shared__ float red[DMODEL];
  const int g = blockIdx.x, tid = threadIdx.x;
  float c = fmaxf(cnt[g], 1.f);
  sp[tid] = sums[g * DMODEL + tid] / c;
  __syncthreads();
  float acc = bm1[tid];
  for (int k = 0; k < DMODEL; ++k) acc += sp[k] * Wm1[tid * DMODEL + k];
  red[tid] = fmaxf(acc, 0.f) * Wm2[tid];
  __syncthreads();
  for (int off = 64; off > 0; off >>= 1) {
    if (tid < off) red[tid] += red[tid + off];
    __syncthreads();
  }
  if (tid == 0) out[g] = red[0] + bm2[0];
}

// ---------------------------------------------------------------------------
extern "C" void kernel_launch(void* const* d_in, const int* in_sizes, int n_in,
                              void* d_out, int out_size, void* d_ws,
                              size_t ws_size, hipStream_t stream) {
  (void)in_sizes; (void)n_in; (void)out_size; (void)ws_size;
  const float* x    = (const float*)d_in[0];
  const int*   bidx = (const int*)d_in[1];
  const float* Wqkv = (const float*)d_in[2];
  const float* bqkv = (const float*)d_in[3];
  const float* Wo   = (const float*)d_in[4];
  const float* bo   = (const float*)d_in[5];
  const float* ln1g = (const float*)d_in[6];
  const float* ln1b = (const float*)d_in[7];
  const float* W1   = (const float*)d_in[8];
  const float* b1   = (const float*)d_in[9];
  const float* W2   = (const float*)d_in[10];
  const float* b2   = (const float*)d_in[11];
  const float* ln2g = (const float*)d_in[12];
  const float* ln2b = (const float*)d_in[13];
  const float* Wm1  = (const float*)d_in[14];
  const float* bm1  = (const float*)d_in[15];
  const float* Wm2  = (const float*)d_in[16];
  const float* bm2  = (const float*)d_in[17];
  float* out = (float*)d_out;

  char* ws = (char*)d_ws;
  size_t off = 0;
  auto alloc = [&](size_t bytes) -> void* {
    void* p = ws + off;
    off += (bytes + 255) & ~(size_t)255;
    return p;
  };
  int*   topk = (int*)  alloc((size_t)N_NODES * TOPK * 4);
  float* qkv  = (float*)alloc((size_t)N_NODES * 384 * 4);
  float* ctx  = (float*)alloc((size_t)N_NODES * DMODEL * 4);
  float* h1   = (float*)alloc((size_t)N_NODES * DMODEL * 4);
  float* ff   = (float*)alloc((size_t)N_NODES * FFDIM * 4);
  float* h2   = (float*)alloc((size_t)N_NODES * DMODEL * 4);
  float* sums = (float*)alloc((size_t)NGRAPH * DMODEL * 4);
  float* cnts = (float*)alloc((size_t)NGRAPH * 4);

  sim_topk_kernel<<<N_NODES / 16, 256, 0, stream>>>(x, topk);

  size_t sm_qkv = (size_t)(16 * (128 + 4) + 16 * (384 + 4)) * 4;
  gemm_ln_kernel<<<N_NODES / 16, 256, sm_qkv, stream>>>(
      x, Wqkv, bqkv, nullptr, nullptr, nullptr, qkv, 128, 384, 0);

  attn_kernel<<<N_NODES, 128, 0, stream>>>(qkv, topk, ctx);

  size_t sm_128 = (size_t)(16 * (128 + 4) + 16 * (128 + 4)) * 4;
  gemm_ln_kernel<<<N_NODES / 16, 256, sm_128, stream>>>(
      ctx, Wo, bo, x, ln1g, ln1b, h1, 128, 128, 0);

  size_t sm_ff1 = (size_t)(16 * (128 + 4) + 16 * (256 + 4)) * 4;
  gemm_ln_kernel<<<N_NODES / 16, 256, sm_ff1, stream>>>(
      h1, W1, b1, nullptr, nullptr, nullptr, ff, 128, 256, 1);

  size_t sm_ff2 = (size_t)(16 * (256 + 4) + 16 * (128 + 4)) * 4;
  gemm_ln_kernel<<<N_NODES / 16, 256, sm_ff2, stream>>>(
      ff, W2, b2, h1, ln2g, ln2b, h2, 256, 128, 0);

  pool_init_kernel<<<(NGRAPH * DMODEL + 255) / 256, 256, 0, stream>>>(sums, cnts);
  pool_kernel<<<N_NODES, 128, 0, stream>>>(h2, bidx, sums, cnts);
  head_kernel<<<NGRAPH, 128, 0, stream>>>(sums, cnts, Wm1, bm1, Wm2, bm2, out);
}